// PointToMeshResidual_11751030522281
// MI455X (gfx1250) — compile-verified
//
#include <hip/hip_runtime.h>

// Problem constants (from reference): B=1, F=8192, Q=2048
#define FQ 8192
#define QQ 2048
#define CH 256   // faces staged per LDS chunk (24 KB of LDS)

typedef __attribute__((ext_vector_type(2))) float v2f;
typedef __attribute__((ext_vector_type(8))) float v8f;

// Safe reciprocal matching jnp.where(|d|<eps, eps, d) guarding, but using the
// single-instruction v_rcp_f32 (1 ulp) instead of the IEEE div expansion.
__device__ __forceinline__ float rcp_safe(float d) {
    d = (fabsf(d) < 1e-12f) ? 1e-12f : d;
    return __builtin_amdgcn_rcpf(d);
}
__device__ __forceinline__ float clamp01(float x) {
    return fminf(fmaxf(x, 0.f), 1.f);
}

// ---------------------------------------------------------------------------
// Kernel 0: per-face precompute.
// faceB row (12 floats): [ab.x,ab.y,ab.z,0, ac.x,ac.y,ac.z,0, a.x,a.y,a.z,0]
//   -> zero pads let the WMMA B-tile build index K-components branchlessly
//      (k1==3 lands on the pad).
// faceK row (12 floats): [ab.a, ac.a, ab.b, ac.b, ab.c, ac.c,
//                         |a|^2, |ab|^2, ab.ac, |ac|^2, pad2]
// ---------------------------------------------------------------------------
__global__ void prep_faces(const float* __restrict__ tris,
                           float* __restrict__ faceB,
                           float* __restrict__ faceK) {
    int f = blockIdx.x * blockDim.x + threadIdx.x;
    if (f >= FQ) return;
    const float* t = tris + f * 9;
    float ax = t[0], ay = t[1], az = t[2];
    float bx = t[3], by = t[4], bz = t[5];
    float cx = t[6], cy = t[7], cz = t[8];
    float abx = bx - ax, aby = by - ay, abz = bz - az;
    float acx = cx - ax, acy = cy - ay, acz = cz - az;
    float* Bp = faceB + f * 12;
    Bp[0] = abx; Bp[1] = aby; Bp[2]  = abz; Bp[3]  = 0.f;
    Bp[4] = acx; Bp[5] = acy; Bp[6]  = acz; Bp[7]  = 0.f;
    Bp[8] = ax;  Bp[9] = ay;  Bp[10] = az;  Bp[11] = 0.f;
    float* Kp = faceK + f * 12;
    Kp[0] = abx*ax + aby*ay + abz*az;
    Kp[1] = acx*ax + acy*ay + acz*az;
    Kp[2] = abx*bx + aby*by + abz*bz;
    Kp[3] = acx*bx + acy*by + acz*bz;
    Kp[4] = abx*cx + aby*cy + abz*cz;
    Kp[5] = acx*cx + acy*cy + acz*cz;
    Kp[6] = ax*ax + ay*ay + az*az;
    Kp[7] = abx*abx + aby*aby + abz*abz;
    Kp[8] = abx*acx + aby*acy + abz*acz;
    Kp[9] = acx*acx + acy*acy + acz*acz;
    Kp[10] = 0.f; Kp[11] = 0.f;
}

// ---------------------------------------------------------------------------
// Kernel 1: brute-force closest triangle per query.
// 8 waves/block, each wave owns one 16-query tile; faces stream through LDS.
// s1 = P·ab^T, s2 = P·ac^T, s3 = P·a^T via V_WMMA_F32_16X16X4_F32 (K padded).
// Region logic is fully branchless (cndmask selects), divisions are v_rcp.
// ---------------------------------------------------------------------------
__global__ __launch_bounds__(256)
void closest_kernel(const float* __restrict__ pts,
                    const float* __restrict__ faceB,
                    const float* __restrict__ faceK,
                    float* __restrict__ outVW,
                    int* __restrict__ outF) {
    __shared__ __align__(16) float ldsB[CH * 12];
    __shared__ __align__(16) float ldsK[CH * 12];

    const int tid  = threadIdx.x;
    const int lane = tid & 31;        // wave32
    const int wave = tid >> 5;
    const int half = lane >> 4;       // 0: K=0,1 ; 1: K=2,3 (A/B VGPR layout)
    const int n    = lane & 15;
    const int k0   = half * 2;        // K component held in VGPR0
    const int k1   = k0 + 1;          // K component held in VGPR1 (3 -> pad 0)
    const int qbase = blockIdx.x * 128 + wave * 16;

    // A matrix (16x4 f32): lane holds query row m=n, K components per half.
    const float* pA = pts + (qbase + n) * 3;
    float px0 = pA[0], px1 = pA[1], px2 = pA[2];
    v2f aP;
    aP.x = (half == 0) ? px0 : px2;
    aP.y = (half == 0) ? px1 : 0.f;

    // |p|^2 for the 8 query rows this lane's C-slots cover.
    float pq2[8];
#pragma unroll
    for (int r = 0; r < 8; ++r) {
        const float* pp = pts + (qbase + r + half * 8) * 3;
        pq2[r] = pp[0]*pp[0] + pp[1]*pp[1] + pp[2]*pp[2];
    }

    float bd[8], bv[8], bw[8];
    int   bfi[8];
#pragma unroll
    for (int r = 0; r < 8; ++r) { bd[r] = 3.4e38f; bv[r] = 0.f; bw[r] = 0.f; bfi[r] = 0; }

    const v8f z = {};

    for (int chunk = 0; chunk < FQ / CH; ++chunk) {
        const int base = chunk * CH;
        // cooperative 128-bit stage of CH faces into LDS
        const float4* gB = (const float4*)(faceB + base * 12);
        const float4* gK = (const float4*)(faceK + base * 12);
        float4* sB = (float4*)ldsB;
        float4* sK = (float4*)ldsK;
        for (int i = tid; i < CH * 3; i += 256) { sB[i] = gB[i]; sK[i] = gK[i]; }
        __syncthreads();
        if (chunk + 1 < FQ / CH) {   // warm L2/WGP$ for the next chunk
            __builtin_prefetch(faceB + (base + CH) * 12, 0, 0);
            __builtin_prefetch(faceK + (base + CH) * 12, 0, 0);
        }

        for (int t = 0; t < CH / 16; ++t) {
            const int fl = t * 16 + n;
            const float* rB = ldsB + fl * 12;
            // Branchless B-tile build: pads at k==3 supply the zero.
            v2f bAB, bAC, bA;
            bAB.x = rB[k0];     bAB.y = rB[k1];
            bAC.x = rB[4 + k0]; bAC.y = rB[4 + k1];
            bA.x  = rB[8 + k0]; bA.y  = rB[8 + k1];

            const float* rK = ldsK + fl * 12;
            const float c1 = rK[0], c2 = rK[1], c3 = rK[2], c4 = rK[3], c5 = rK[4];
            const float c6 = rK[5], c7 = rK[6], c8 = rK[7], c9 = rK[8], c10 = rK[9];

            v8f S1 = __builtin_amdgcn_wmma_f32_16x16x4_f32(false, aP, false, bAB, (short)0, z, false, false);
            v8f S2 = __builtin_amdgcn_wmma_f32_16x16x4_f32(false, aP, false, bAC, (short)0, z, false, false);
            v8f S3 = __builtin_amdgcn_wmma_f32_16x16x4_f32(false, aP, false, bA,  (short)0, z, false, false);

            const int f = base + t * 16 + n;
#pragma unroll
            for (int r = 0; r < 8; ++r) {
                float s1 = S1[r], s2 = S2[r], s3 = S3[r];
                float d1 = s1 - c1, d2 = s2 - c2;
                float d3 = s1 - c3, d4 = s2 - c4;
                float d5 = s1 - c5, d6 = s2 - c6;
                float vc = d1*d4 - d3*d2;
                float vb = d5*d2 - d1*d6;
                float va = d3*d6 - d5*d4;

                // Interior point
                float rden = rcp_safe(va + vb + vc);
                float v = vb * rden;
                float w = vc * rden;
                // Edge BC
                float e43 = d4 - d3, e56 = d5 - d6;
                float wbc = e43 * rcp_safe(e43 + e56);
                bool cbc = (va <= 0.f) && (e43 >= 0.f) && (e56 >= 0.f);
                v = cbc ? 1.f - wbc : v;  w = cbc ? wbc : w;
                // Edge AC
                float wac = d2 * rcp_safe(d2 - d6);
                bool cac = (vb <= 0.f) && (d2 >= 0.f) && (d6 <= 0.f);
                v = cac ? 0.f : v;        w = cac ? wac : w;
                // Edge AB
                float vab = d1 * rcp_safe(d1 - d3);
                bool cab = (vc <= 0.f) && (d1 >= 0.f) && (d3 <= 0.f);
                v = cab ? vab : v;        w = cab ? 0.f : w;
                // Vertices C, B, A (later wins, matching reference order)
                bool cC = (d6 >= 0.f) && (d5 <= d6);
                v = cC ? 0.f : v;         w = cC ? 1.f : w;
                bool cB = (d3 >= 0.f) && (d4 <= d3);
                v = cB ? 1.f : v;         w = cB ? 0.f : w;
                bool cA = (d1 <= 0.f) && (d2 <= 0.f);
                v = cA ? 0.f : v;         w = cA ? 0.f : w;

                float ap2 = pq2[r] - 2.f * s3 + c7;
                float dist2 = ap2 - 2.f*v*d1 - 2.f*w*d2
                            + v*v*c8 + 2.f*v*w*c9 + w*w*c10;
                bool better = dist2 < bd[r];
                bd[r]  = better ? dist2 : bd[r];
                bv[r]  = better ? v     : bv[r];
                bw[r]  = better ? w     : bw[r];
                bfi[r] = better ? f     : bfi[r];
            }
        }
        __syncthreads();
    }

    // 16-lane butterfly argmin per half-wave; tie-break on lowest face id.
#pragma unroll
    for (int r = 0; r < 8; ++r) {
        float d = bd[r], v = bv[r], w = bw[r];
        int   f = bfi[r];
#pragma unroll
        for (int m = 8; m >= 1; m >>= 1) {
            float od = __shfl_xor(d, m, 32);
            int   of = __shfl_xor(f, m, 32);
            float ov = __shfl_xor(v, m, 32);
            float ow = __shfl_xor(w, m, 32);
            bool take = (od < d) || (od == d && of < f);
            d = take ? od : d;
            f = take ? of : f;
            v = take ? ov : v;
            w = take ? ow : w;
        }
        if (n == 0) {
            int q = qbase + half * 8 + r;
            outVW[q * 2 + 0] = v;
            outVW[q * 2 + 1] = w;
            outF[q] = f;
        }
    }
}

// ---------------------------------------------------------------------------
// Kernel 2: gather + interpolate + argmax. Output layout (floats):
// [0,3Q) residual | [3Q,6Q) normals | [6Q,9Q) cmaps | [9Q,10Q) closest_idx
// ---------------------------------------------------------------------------
__global__ void gather_kernel(const float* __restrict__ tris,
                              const float* __restrict__ pts,
                              const float* __restrict__ nrms,
                              const float* __restrict__ cmps,
                              const int* __restrict__ faces,
                              const float* __restrict__ bcsVW,
                              const int* __restrict__ bfid,
                              float* __restrict__ out) {
    int q = blockIdx.x * blockDim.x + threadIdx.x;
    if (q >= QQ) return;
    int f = bfid[q];
    float v = bcsVW[q * 2 + 0];
    float w = bcsVW[q * 2 + 1];
    float u = 1.f - v - w;
    u = clamp01(u); v = clamp01(v); w = clamp01(w);   // reference clips each bc

    const float* T = tris + f * 9;
    const float* N = nrms + f * 9;
    const float* C = cmps + f * 9;
    float px = pts[q*3+0], py = pts[q*3+1], pz = pts[q*3+2];

#pragma unroll
    for (int d = 0; d < 3; ++d) {
        float ti = u*T[d] + v*T[3+d] + w*T[6+d];
        float ni = u*N[d] + v*N[3+d] + w*N[6+d];
        float ci = u*C[d] + v*C[3+d] + w*C[6+d];
        float pd = (d == 0) ? px : (d == 1) ? py : pz;
        out[            q*3 + d] = ti - pd;   // residual
        out[3*QQ      + q*3 + d] = ni;        // closest_normals
        out[6*QQ      + q*3 + d] = ci;        // closest_cmaps
    }
    // argmax of clipped bcs (first max wins, like jnp.argmax)
    int vmax = 0; float bb = u;
    if (v > bb) { vmax = 1; bb = v; }
    if (w > bb) { vmax = 2; }
    out[9*QQ + q] = (float)faces[f*3 + vmax]; // closest_idx (exact in fp32)
}

// ---------------------------------------------------------------------------
extern "C" void kernel_launch(void* const* d_in, const int* in_sizes, int n_in,
                              void* d_out, int out_size, void* d_ws, size_t ws_size,
                              hipStream_t stream) {
    const float* tris  = (const float*)d_in[0];  // [1,F,3,3]
    const float* pts   = (const float*)d_in[1];  // [1,Q,3]
    const float* nrms  = (const float*)d_in[2];  // [1,F,3,3]
    const float* cmps  = (const float*)d_in[3];  // [1,F,3,3]
    const int*   faces = (const int*)d_in[4];    // [1,F,3]

    float* faceB = (float*)d_ws;                 // F*12 floats
    float* faceK = faceB + FQ * 12;              // F*12 floats
    float* bcsVW = faceK + FQ * 12;              // Q*2 floats
    int*   bfid  = (int*)(bcsVW + QQ * 2);       // Q ints
    float* out   = (float*)d_out;

    prep_faces<<<FQ / 256, 256, 0, stream>>>(tris, faceB, faceK);
    closest_kernel<<<QQ / 128, 256, 0, stream>>>(pts, faceB, faceK, bcsVW, bfid);
    gather_kernel<<<QQ / 256, 256, 0, stream>>>(tris, pts, nrms, cmps, faces,
                                                bcsVW, bfid, out);
}